// Izhikevich_SNN_13752485281933
// MI455X (gfx1250) — compile-verified
//
#include <hip/hip_runtime.h>

typedef float v2f __attribute__((ext_vector_type(2)));
typedef float v8f __attribute__((ext_vector_type(8)));

#define T_STEPS 500
#define B_SZ    256
#define N_SZ    256
#define BN      (B_SZ * N_SZ)

// ---------------------------------------------------------------------------
// Kernel 1: fused feedforward GEMM (fp32 WMMA) + Izhikevich neuron scan.
// Grid: (B/16)*(N/16) = 256 blocks, 1 wave (32 threads) each.
// Each wave owns a 16(batch) x 16(neuron) tile and carries v,u in registers
// across the T=500 timestep loop. W1 tile (16x256) is loop-invariant and is
// kept in VGPRs by the compiler; x is streamed from L2 (entire 131 MB input
// fits in the 192 MB L2). All output stores are non-temporal so the 525 MB
// result stream does not evict the reused x from L2.
//
// WMMA f32 16x16x4 operand layout (ISA 7.12.2, 32-bit):
//   A (16x4, MxK):  lane L<16: M=L,    v0=K0, v1=K1 ; lane>=16: M=L-16, v0=K2, v1=K3
//   B (4x16, KxN):  lane L<16: N=L,    v0=K0, v1=K1 ; lane>=16: N=L-16, v0=K2, v1=K3
//   C/D (16x16):    reg r, lane L: M = r + 8*(L/16), N = L%16
// ---------------------------------------------------------------------------
__global__ __launch_bounds__(32)
void snn_scan_kernel(const float* __restrict__ x,    // [B,T,N]
                     const float* __restrict__ st0,  // [3,B,N] (v,u,s)
                     const float* __restrict__ W1,   // [N,N]
                     const float* __restrict__ pa,
                     const float* __restrict__ pb,
                     const float* __restrict__ pc,
                     const float* __restrict__ pd,
                     const float* __restrict__ pv0,
                     const float* __restrict__ pv1,
                     const float* __restrict__ pv2,
                     const float* __restrict__ ptau,
                     const float* __restrict__ pth,
                     float* __restrict__ out_s,       // [T,B,N]
                     float* __restrict__ out_states)  // [T,3,B,N]
{
    const int tile = blockIdx.x;
    const int b0 = (tile / (N_SZ / 16)) * 16;
    const int n0 = (tile % (N_SZ / 16)) * 16;
    const int lane = threadIdx.x;
    const int half = lane >> 4;   // 0: lanes 0-15, 1: lanes 16-31
    const int lid  = lane & 15;
    const int koff = half * 2;    // per-lane K offset inside each K=4 step

    // Per-lane neuron parameters (N index == lid for every C reg).
    const int gn = n0 + lid;
    const float P_a  = pa[gn];
    const float P_b  = pb[gn];
    const float P_c  = pc[gn];
    const float P_d  = pd[gn];
    const float P_v0 = pv0[gn];
    const float P_v1 = pv1[gn];
    const float P_v2 = pv2[gn];
    const float P_k1 = P_a / ptau[gn];   // DT/tau_u * a, DT = 1
    const float P_th = pth[gn];

    // Carried state: reg r corresponds to global batch gb = b0 + half*8 + r.
    float v[8], u[8];
#pragma unroll
    for (int r = 0; r < 8; ++r) {
        const int gb = b0 + half * 8 + r;
        v[r] = st0[0 * BN + gb * N_SZ + gn];
        u[r] = st0[1 * BN + gb * N_SZ + gn];
    }

    // A operand row: x[b0+lid, t, :]  (x layout [B,T,N])
    const float* __restrict__ xrow = x + (size_t)(b0 + lid) * T_STEPS * N_SZ;
    // B operand row: W1[n0+lid, :]   (B[k][n] = W1[n][k]); loop-invariant,
    // hoisted into VGPRs by the compiler.
    const float* __restrict__ wrow = W1 + (size_t)(n0 + lid) * N_SZ + koff;

    for (int t = 0; t < T_STEPS; ++t) {
        const float* __restrict__ xr = xrow + (size_t)t * N_SZ + koff;

        // 4 independent accumulator chains (hide WMMA RAW latency).
        v8f a0 = {}, a1 = {}, a2 = {}, a3 = {};
#pragma unroll
        for (int k = 0; k < N_SZ; k += 16) {
            v2f xa0 = *(const v2f*)(xr + k);
            v2f wb0 = *(const v2f*)(wrow + k);
            a0 = __builtin_amdgcn_wmma_f32_16x16x4_f32(false, xa0, false, wb0,
                                                       (short)0, a0, false, false);
            v2f xa1 = *(const v2f*)(xr + k + 4);
            v2f wb1 = *(const v2f*)(wrow + k + 4);
            a1 = __builtin_amdgcn_wmma_f32_16x16x4_f32(false, xa1, false, wb1,
                                                       (short)0, a1, false, false);
            v2f xa2 = *(const v2f*)(xr + k + 8);
            v2f wb2 = *(const v2f*)(wrow + k + 8);
            a2 = __builtin_amdgcn_wmma_f32_16x16x4_f32(false, xa2, false, wb2,
                                                       (short)0, a2, false, false);
            v2f xa3 = *(const v2f*)(xr + k + 12);
            v2f wb3 = *(const v2f*)(wrow + k + 12);
            a3 = __builtin_amdgcn_wmma_f32_16x16x4_f32(false, xa3, false, wb3,
                                                       (short)0, a3, false, false);
        }
        v8f I = (a0 + a1) + (a2 + a3);

        float* __restrict__ s_t  = out_s + (size_t)t * BN;
        float* __restrict__ st_t = out_states + (size_t)t * 3 * BN;

#pragma unroll
        for (int r = 0; r < 8; ++r) {
            const int gb = b0 + half * 8 + r;
            const size_t idx = (size_t)gb * N_SZ + gn;

            const float vv = v[r];
            const float uu = u[r];
            const float v_mid = vv + (P_v2 * vv * vv + P_v1 * vv + P_v0 - uu + I[r]);
            const float u_mid = uu + P_k1 * (P_b * vv - uu);
            const float s = (v_mid - P_th) > 0.0f ? 1.0f : 0.0f;
            const float v_new = v_mid * (1.0f - s) + s * P_c;
            const float u_new = u_mid + s * P_d;
            v[r] = v_new;
            u[r] = u_new;

            // Non-temporal: streamed outputs must not evict x from L2.
            __builtin_nontemporal_store(s,     s_t + idx);            // outputs [T,B,N]
            __builtin_nontemporal_store(v_new, st_t + 0 * BN + idx);  // states  [T,0,B,N]
            __builtin_nontemporal_store(u_new, st_t + 1 * BN + idx);  // states  [T,1,B,N]
            __builtin_nontemporal_store(s,     st_t + 2 * BN + idx);  // states  [T,2,B,N]
        }
    }
}

// ---------------------------------------------------------------------------
// Kernel 2: leaky-integrator readout. One wave per batch row.
// decoded[t,b] = leak*li + (1-leak)*dot(s[t,b,:], W2)
// Reads the spike tensor back (HBM-rate, ~6 us) and carries the li scan in
// lane 0 after a wave32 butterfly reduction.
// ---------------------------------------------------------------------------
__global__ __launch_bounds__(32)
void li_readout_kernel(const float* __restrict__ out_s,  // [T,B,N]
                       const float* __restrict__ W2,     // [1,N]
                       const float* __restrict__ li0,    // [B,1]
                       const float* __restrict__ leak_p, // [1]
                       float* __restrict__ decoded)      // [T,B]
{
    const int b = blockIdx.x;
    const int lane = threadIdx.x;

    float w[8];
#pragma unroll
    for (int j = 0; j < 8; ++j) w[j] = W2[lane + 32 * j];

    const float leak = leak_p[0];
    float li = li0[b];

    for (int t = 0; t < T_STEPS; ++t) {
        const float* __restrict__ row = out_s + (size_t)t * BN + (size_t)b * N_SZ;
        float p = 0.0f;
#pragma unroll
        for (int j = 0; j < 8; ++j) p += row[lane + 32 * j] * w[j];
        // wave32 butterfly reduction: all lanes end up with the full sum
#pragma unroll
        for (int off = 16; off > 0; off >>= 1) p += __shfl_xor(p, off, 32);
        li = leak * li + (1.0f - leak) * p;
        if (lane == 0) decoded[t * B_SZ + b] = li;
    }
}

extern "C" void kernel_launch(void* const* d_in, const int* in_sizes, int n_in,
                              void* d_out, int out_size, void* d_ws, size_t ws_size,
                              hipStream_t stream) {
    (void)in_sizes; (void)n_in; (void)d_ws; (void)ws_size; (void)out_size;

    const float* x    = (const float*)d_in[0];   // input_batch [B,T,N]
    const float* st   = (const float*)d_in[1];   // state_snn   [3,B,N]
    const float* li0  = (const float*)d_in[2];   // state_LI    [B,1]
    const float* W1   = (const float*)d_in[3];   // [N,N]
    const float* W2   = (const float*)d_in[4];   // [1,N]
    const float* pa   = (const float*)d_in[5];
    const float* pb   = (const float*)d_in[6];
    const float* pc   = (const float*)d_in[7];
    const float* pd   = (const float*)d_in[8];
    const float* pv0  = (const float*)d_in[9];
    const float* pv1  = (const float*)d_in[10];
    const float* pv2  = (const float*)d_in[11];
    const float* ptau = (const float*)d_in[12];
    const float* pth  = (const float*)d_in[13];
    const float* leak = (const float*)d_in[14];

    float* out        = (float*)d_out;
    float* out_s      = out;                            // [T,B,N]
    float* out_states = out + (size_t)T_STEPS * BN;     // [T,3,B,N]
    float* out_dec    = out + (size_t)T_STEPS * BN * 4; // [T,B]

    snn_scan_kernel<<<dim3((B_SZ / 16) * (N_SZ / 16)), dim3(32), 0, stream>>>(
        x, st, W1, pa, pb, pc, pd, pv0, pv1, pv2, ptau, pth, out_s, out_states);

    li_readout_kernel<<<dim3(B_SZ), dim3(32), 0, stream>>>(
        out_s, W2, li0, leak, out_dec);
}